// ProposalGenerator_2482491097552
// MI455X (gfx1250) — compile-verified
//
#include <hip/hip_runtime.h>
#include <stdint.h>

// ---------------------------------------------------------------------------
// ProposalGenerator for MI455X (gfx1250, wave32).
// decode -> radix-select top-2000 -> LDS bitonic sort -> IoU suppression
// bitmask -> single-wave greedy NMS scan (register bitmap + async-DMA mask
// tiles) -> compact top-300.
// No matmul exists in this op (IoU is min/max algebra), so no WMMA; the CDNA5
// features that apply: global_load_async_to_lds_b128 / s_wait_asynccnt
// (double-buffered mask tiling), global_prefetch_b8, wave32 shfl broadcast,
// single-wave workgroups (barriers lower to S_NOP per ISA).
// ---------------------------------------------------------------------------

#define NBATCH   8
#define PRE_NMS  2000
#define MSLOTS   2048        // PRE_NMS padded to pow2 for bitonic sort
#define MWORDS   64          // MSLOTS / 32
#define NTILES   ((PRE_NMS + 31) / 32)   // 63 mask tiles of 32 rows
#define POST_NMS 300
#define NMS_THR  0.7f
#define IMG_DIM  1024.0f
#define MIN_SIZE 16.0f

#if defined(__has_builtin)
#  if __has_builtin(__builtin_amdgcn_global_load_async_to_lds_b128) && \
      __has_builtin(__builtin_amdgcn_s_wait_asynccnt)
#    define HAVE_ASYNC 1
#  endif
#endif

// Exact pointee type the async-LDS builtin expects (from hipcc diagnostic):
// int __attribute__((vector_size(16))) in AS1 (src) / AS3 (dst).
typedef int v4i __attribute__((vector_size(16)));
typedef __attribute__((address_space(1))) v4i* g_v4i_ptr;
typedef __attribute__((address_space(3))) v4i* l_v4i_ptr;

// Order-preserving float->u32 key (larger key == larger float).
__device__ __forceinline__ unsigned ord_key(float f) {
  unsigned u = __float_as_uint(f);
  return (u & 0x80000000u) ? ~u : (u | 0x80000000u);
}
__device__ __forceinline__ float inv_key(unsigned k) {
  unsigned u = (k & 0x80000000u) ? (k ^ 0x80000000u) : ~k;
  return __uint_as_float(u);
}

// ---------------------------------------------------------------------------
// Stage 1: decode anchors + deltas -> clipped boxes, validity-masked score key
// ---------------------------------------------------------------------------
__global__ __launch_bounds__(256) void decode_kernel(
    const float* __restrict__ anchor, const float* __restrict__ reg,
    const float* __restrict__ score, float* __restrict__ boxes,
    unsigned* __restrict__ keys, int total) {
  int i = blockIdx.x * blockDim.x + threadIdx.x;
  if (i >= total) return;
  const float4 a = ((const float4*)anchor)[i];   // y1,x1,y2,x2
  const float4 r = ((const float4*)reg)[i];      // dy,dx,dh,dw
  float ah = a.z - a.x, aw = a.w - a.y;
  float acy = a.x + 0.5f * ah, acx = a.y + 0.5f * aw;
  float cy = r.x * ah + acy, cx = r.y * aw + acx;
  float h = expf(r.z) * ah, w = expf(r.w) * aw;
  float y1 = fminf(fmaxf(cy - 0.5f * h, 0.f), IMG_DIM);
  float y2 = fminf(fmaxf(cy + 0.5f * h, 0.f), IMG_DIM);
  float x1 = fminf(fmaxf(cx - 0.5f * w, 0.f), IMG_DIM);
  float x2 = fminf(fmaxf(cx + 0.5f * w, 0.f), IMG_DIM);
  bool valid = ((y2 - y1) >= MIN_SIZE) && ((x2 - x1) >= MIN_SIZE);
  float s = valid ? score[i] : -INFINITY;
  ((float4*)boxes)[i] = make_float4(y1, x1, y2, x2);
  keys[i] = ord_key(s);
}

// ---------------------------------------------------------------------------
// Stage 2: per-image radix-select top-2000 keys, then LDS bitonic sort
// (descending score, stable by index via composite key | ~idx).
// ---------------------------------------------------------------------------
__global__ __launch_bounds__(1024) void select_sort_kernel(
    const unsigned* __restrict__ keys, const float* __restrict__ boxes,
    float* __restrict__ sel_boxes, float* __restrict__ sel_scores, int N) {
  const int b = blockIdx.x;
  const int tid = threadIdx.x;
  const unsigned* k = keys + (size_t)b * N;

  __shared__ unsigned hist[256];
  __shared__ unsigned s_bin, s_above;
  __shared__ unsigned cnt_gt, cnt_eq;
  __shared__ unsigned long long sel[MSLOTS];  // 16 KB of 320 KB LDS

  // ---- 4-pass radix select: find T = 2000th-largest key ----
  unsigned prefix = 0;
  int remK = PRE_NMS;   // ties at T still needed
  int G = 0;            // count strictly greater than T
  for (int pass = 3; pass >= 0; --pass) {
    int shift = pass * 8;
    if (tid < 256) hist[tid] = 0;
    __syncthreads();
    for (int n = tid; n < N; n += 1024) {
      unsigned key = k[n];
      if (pass == 3 || (key >> (shift + 8)) == prefix)
        atomicAdd(&hist[(key >> shift) & 255u], 1u);
    }
    __syncthreads();
    if (tid == 0) {
      unsigned acc = 0;
      int bin = 255;
      for (; bin > 0; --bin) {
        if (acc + hist[bin] >= (unsigned)remK) break;
        acc += hist[bin];
      }
      s_bin = (unsigned)bin;
      s_above = acc;
    }
    __syncthreads();
    prefix = (prefix << 8) | s_bin;
    G += (int)s_above;
    remK -= (int)s_above;
    __syncthreads();
  }
  const unsigned T = prefix;

  // ---- compaction into LDS: strictly-greater first, then remK ties ----
  if (tid == 0) { cnt_gt = 0; cnt_eq = 0; }
  for (int i = tid; i < MSLOTS; i += 1024) sel[i] = 0ull;  // sentinel (sorts last)
  __syncthreads();
  for (int n = tid; n < N; n += 1024) {
    unsigned key = k[n];
    if (key > T) {
      unsigned p = atomicAdd(&cnt_gt, 1u);
      if (p < (unsigned)PRE_NMS)
        sel[p] = ((unsigned long long)key << 32) | (unsigned)~n;
    }
  }
  __syncthreads();
  for (int n = tid; n < N; n += 1024) {
    unsigned key = k[n];
    if (key == T) {
      unsigned p = atomicAdd(&cnt_eq, 1u);
      if ((int)p < remK)
        sel[G + (int)p] = ((unsigned long long)key << 32) | (unsigned)~n;
    }
  }
  __syncthreads();

  // ---- bitonic sort, 2048 elems, descending (key desc, idx asc via ~idx) ----
  for (unsigned kk = 2; kk <= MSLOTS; kk <<= 1) {
    for (unsigned j = kk >> 1; j > 0; j >>= 1) {
      for (unsigned i = tid; i < MSLOTS; i += 1024) {
        unsigned ixj = i ^ j;
        if (ixj > i) {
          bool up = (i & kk) != 0;  // inverted -> overall descending
          unsigned long long x = sel[i], y = sel[ixj];
          if ((x > y) == up) { sel[i] = y; sel[ixj] = x; }
        }
      }
      __syncthreads();
    }
  }

  // ---- emit sorted selection: boxes + raw scores (-inf marks dead) ----
  for (int i = tid; i < MSLOTS; i += 1024) {
    unsigned long long e = sel[i];
    unsigned key = (unsigned)(e >> 32);
    unsigned idx = ~(unsigned)e;
    float4 bx = make_float4(0.f, 0.f, 0.f, 0.f);
    float s = -INFINITY;
    if (i < PRE_NMS && idx < (unsigned)N && key != 0u) {
      bx = ((const float4*)boxes)[(size_t)b * N + idx];
      s = inv_key(key);
    }
    ((float4*)sel_boxes)[(size_t)b * MSLOTS + i] = bx;
    sel_scores[(size_t)b * MSLOTS + i] = s;
  }
}

// ---------------------------------------------------------------------------
// Stage 3: suppression bitmask. Block = 256 rows x one 32-column word.
// Column boxes staged into LDS via CDNA5 async global->LDS DMA (ASYNCcnt).
// ---------------------------------------------------------------------------
__global__ __launch_bounds__(256) void iou_mask_kernel(
    const float* __restrict__ sel_boxes, unsigned* __restrict__ mask) {
  const int word = blockIdx.x;              // 0..63
  const int row = blockIdx.y * 256 + threadIdx.x;
  const int b = blockIdx.z;
  const int tid = threadIdx.x;

  __shared__ __align__(16) float4 colb4[32];
  float* colb = (float*)colb4;
  const float* base = sel_boxes + ((size_t)b * MSLOTS + (size_t)word * 32) * 4;

#if defined(HAVE_ASYNC)
  if (tid < 32) {
    // one b128 per lane: 32 lanes x 16B = 512B column tile, no VGPR staging
    __builtin_amdgcn_global_load_async_to_lds_b128(
        (g_v4i_ptr)(base + tid * 4),
        (l_v4i_ptr)(&colb4[tid]), 0, 0);
  }
  __builtin_amdgcn_s_wait_asynccnt(0);
#else
  if (tid < 128) colb[tid] = base[tid];
#endif
  __syncthreads();

  float4 rb = ((const float4*)sel_boxes)[(size_t)b * MSLOTS + row];
  float rArea = (rb.z - rb.x) * (rb.w - rb.y);
  unsigned bits = 0;
#pragma unroll
  for (int c = 0; c < 32; ++c) {
    float cy1 = colb[c * 4 + 0], cx1 = colb[c * 4 + 1];
    float cy2 = colb[c * 4 + 2], cx2 = colb[c * 4 + 3];
    float ty = fmaxf(rb.x, cy1);
    float by = fminf(rb.z, cy2);
    float lx = fmaxf(rb.y, cx1);
    float rx = fminf(rb.w, cx2);
    float inter = fmaxf(by - ty, 0.f) * fmaxf(rx - lx, 0.f);
    float cArea = (cy2 - cy1) * (cx2 - cx1);
    float iou = inter / (rArea + cArea - inter + 1e-9f);
    int j = word * 32 + c;
    if ((iou > NMS_THR) && (j > row)) bits |= (1u << c);
  }
  mask[((size_t)b * MSLOTS + row) * MWORDS + word] = bits;
}

// ---------------------------------------------------------------------------
// Stage 4 helper: stage one 32-row x 64-word mask tile (8 KB) into LDS.
// Async path: 16 b128 DMA loads per lane, tracked on ASYNCcnt.
// ---------------------------------------------------------------------------
__device__ __forceinline__ void load_tile(const unsigned* __restrict__ g,
                                          unsigned* dst, int t, int tid) {
  const unsigned* src = g + (size_t)t * (32 * MWORDS);
#if defined(HAVE_ASYNC)
#pragma unroll
  for (int c = 0; c < 16; ++c) {
    int w4 = (c * 32 + tid) * 4;        // 16B-aligned chunk per lane
    __builtin_amdgcn_global_load_async_to_lds_b128(
        (g_v4i_ptr)(src + w4), (l_v4i_ptr)(dst + w4), 0, 0);
  }
#else
  for (int c = tid; c < 32 * MWORDS; c += 32) dst[c] = src[c];
#endif
}

// ---------------------------------------------------------------------------
// Stage 4: greedy NMS scan + compaction. ONE WAVE per image. The 2048-bit
// removal bitmap lives in registers (2 words/lane); the test word is
// broadcast with wave32 shfl, so the serial chain is pure register dataflow.
// Mask rows stream through a double-buffered async-DMA LDS tile:
// s_wait_asynccnt(16) <=> previous tile landed (ASYNCcnt retires in order).
// ---------------------------------------------------------------------------
__global__ __launch_bounds__(32) void nms_scan_kernel(
    const float* __restrict__ sel_boxes, const float* __restrict__ sel_scores,
    const unsigned* __restrict__ mask, float* __restrict__ out_boxes,
    float* __restrict__ out_scores, float* __restrict__ out_idx) {
  const int b = blockIdx.x;
  const int tid = threadIdx.x;
  __shared__ __align__(16) unsigned tile[2][32 * MWORDS];  // 2 x 8 KB
  __shared__ unsigned kept[MWORDS];
  __shared__ unsigned pfx[MWORDS];

  // removal bitmap in registers: lane owns words tid (r0) and tid+32 (r1);
  // bit set = dead (not alive0: non-finite score or padding slot)
  unsigned r0 = 0, r1 = 0;
#pragma unroll 4
  for (int bit = 0; bit < 32; ++bit) {
    int s0 = tid * 32 + bit;
    int s1 = (tid + 32) * 32 + bit;
    unsigned u0 = __float_as_uint(sel_scores[(size_t)b * MSLOTS + s0]);
    unsigned u1 = __float_as_uint(sel_scores[(size_t)b * MSLOTS + s1]);
    if ((u0 & 0x7F800000u) == 0x7F800000u || s0 >= PRE_NMS) r0 |= 1u << bit;
    if ((u1 & 0x7F800000u) == 0x7F800000u || s1 >= PRE_NMS) r1 |= 1u << bit;
  }

  const unsigned* g = mask + (size_t)b * MSLOTS * MWORDS;
  load_tile(g, tile[0], 0, tid);

  for (int t = 0; t < NTILES; ++t) {
    if (t + 1 < NTILES) {
      load_tile(g, tile[(t + 1) & 1], t + 1, tid);   // overlap DMA with compute
      if (tid == 0 && t + 2 < NTILES)                // depth-2 L2 prefetch
        __builtin_prefetch(g + (size_t)(t + 2) * (32 * MWORDS), 0, 1);
#if defined(HAVE_ASYNC)
      __builtin_amdgcn_s_wait_asynccnt(16);          // tile t complete
#endif
    } else {
#if defined(HAVE_ASYNC)
      __builtin_amdgcn_s_wait_asynccnt(0);
#endif
    }
    __threadfence_block();   // keep LDS tile reads below the wait

    const unsigned* trow = tile[t & 1];
    int rows = PRE_NMS - t * 32;
    if (rows > 32) rows = 32;
    for (int k2 = 0; k2 < rows; ++k2) {
      int i = t * 32 + k2;
      int w = i >> 5;                         // uniform across lanes
      unsigned rw = (w < 32) ? (unsigned)__shfl((int)r0, w & 31, 32)
                             : (unsigned)__shfl((int)r1, w & 31, 32);
      if (!((rw >> (i & 31)) & 1u)) {         // candidate i still alive
        r0 |= trow[k2 * MWORDS + tid];
        r1 |= trow[k2 * MWORDS + tid + 32];
      }
    }
  }

  // survivors bitmap + serial prefix popcount (64 words, trivial)
  kept[tid] = ~r0;
  kept[tid + 32] = ~r1;
  __syncthreads();   // S_NOP: single-wave workgroup (compiler fence only)
  if (tid == 0) {
    unsigned acc = 0;
    for (int w = 0; w < MWORDS; ++w) { pfx[w] = acc; acc += __popc(kept[w]); }
  }
  __syncthreads();

  // zero-pad outputs; indices are always the batch id (as float)
  for (int j = tid; j < POST_NMS; j += 32) {
    out_scores[b * POST_NMS + j] = 0.f;
    out_idx[b * POST_NMS + j] = (float)b;
    ((float4*)out_boxes)[b * POST_NMS + j] = make_float4(0.f, 0.f, 0.f, 0.f);
  }
  __syncthreads();

  // scatter survivors (already score-sorted) into first POST_NMS slots
#pragma unroll
  for (int h = 0; h < 2; ++h) {
    int w = tid + 32 * h;
    unsigned bits = kept[w];
    int r = (int)pfx[w];
    while (bits) {
      int bit = __builtin_ctz(bits);
      bits &= bits - 1;
      if (r < POST_NMS) {
        int slot = w * 32 + bit;
        ((float4*)out_boxes)[b * POST_NMS + r] =
            ((const float4*)sel_boxes)[(size_t)b * MSLOTS + slot];
        out_scores[b * POST_NMS + r] = sel_scores[(size_t)b * MSLOTS + slot];
      }
      ++r;
    }
  }
}

// ---------------------------------------------------------------------------
extern "C" void kernel_launch(void* const* d_in, const int* in_sizes, int n_in,
                              void* d_out, int out_size, void* d_ws, size_t ws_size,
                              hipStream_t stream) {
  const float* anchor = (const float*)d_in[0];   // [B,N,4]
  const float* boxreg = (const float*)d_in[1];   // [B,N,4]
  const float* obj    = (const float*)d_in[2];   // [B,N]
  const int N = in_sizes[2] / NBATCH;            // 147456
  const int total = NBATCH * N;

  // Workspace carve-up (~28 MB total, 256B-aligned slabs):
  char* ws = (char*)d_ws;
  size_t o = 0;
  float* boxes = (float*)(ws + o);
  o += ((size_t)total * 4 * sizeof(float) + 255) & ~(size_t)255;      // 18.9 MB
  unsigned* keys = (unsigned*)(ws + o);
  o += ((size_t)total * sizeof(unsigned) + 255) & ~(size_t)255;       //  4.7 MB
  float* sel_boxes = (float*)(ws + o);
  o += ((size_t)NBATCH * MSLOTS * 4 * sizeof(float) + 255) & ~(size_t)255;
  float* sel_scores = (float*)(ws + o);
  o += ((size_t)NBATCH * MSLOTS * sizeof(float) + 255) & ~(size_t)255;
  unsigned* mask = (unsigned*)(ws + o);                               //  4.0 MB
  (void)ws_size; (void)n_in; (void)out_size;

  float* out_boxes  = (float*)d_out;                       // [B,300,4]
  float* out_scores = out_boxes + NBATCH * POST_NMS * 4;   // [B,300]
  float* out_idx    = out_scores + NBATCH * POST_NMS;      // [B,300]

  decode_kernel<<<(total + 255) / 256, 256, 0, stream>>>(
      anchor, boxreg, obj, boxes, keys, total);
  select_sort_kernel<<<NBATCH, 1024, 0, stream>>>(
      keys, boxes, sel_boxes, sel_scores, N);
  iou_mask_kernel<<<dim3(MWORDS, MSLOTS / 256, NBATCH), 256, 0, stream>>>(
      sel_boxes, mask);
  nms_scan_kernel<<<NBATCH, 32, 0, stream>>>(
      sel_boxes, sel_scores, mask, out_boxes, out_scores, out_idx);
}